// PatchSelect_52982716563772
// MI455X (gfx1250) — compile-verified
//
#include <hip/hip_runtime.h>
#include <hip/hip_bf16.h>

typedef float v8f __attribute__((ext_vector_type(8)));
typedef float v2f __attribute__((ext_vector_type(2)));

#define BATCH 4
#define CCH   32
#define HH    64
#define WW    64
#define KWW   192
#define PP    129
#define NPT   9          // ceil(129/16) shift tiles of 16
#define NFC   32         // feature chunks
#define FEAT  (CCH*HH*WW) // 131072

// Kernel 1: per-(batch, shift-tile, feature-chunk) partial L1 sums.
// 8 waves/WG. Wave layout: M (shift) = lane%16, K (sub-chunk) = (lane/16)*2 + j,
// which is exactly the V_WMMA_F32_16X16X4_F32 A-matrix layout, so the K-reduction
// is a single WMMA against a ones B-matrix.
__global__ __launch_bounds__(256)
void PatchSelect_partial_kernel(const float* __restrict__ query,
                                const float* __restrict__ key,
                                float* __restrict__ partial) {
  const int wg = blockIdx.x;
  const int fc = wg % NFC;
  const int pt = (wg / NFC) % NPT;
  const int b  = wg / (NFC * NPT);

  const int tid   = threadIdx.x;
  const int wave  = tid >> 5;
  const int lane  = tid & 31;
  const int m     = lane & 15;       // shift within tile -> WMMA A row
  const int khalf = lane >> 4;       // K half -> WMMA A lanes 16-31 hold K=2,3
  const int p     = min(pt * 16 + m, PP - 1);   // clamp keeps key reads in bounds

  const float* qb = query + b * FEAT;
  const float* kb = key   + b * (CCH * HH * KWW);

  float acc[2] = {0.0f, 0.0f};
#pragma unroll
  for (int j = 0; j < 2; ++j) {
    const int sub = wave * 4 + khalf * 2 + j;   // 0..31 sub-chunks of 128 feats
    const int f0  = fc * 4096 + sub * 128;      // flat (c,h,w) start, 2 full rows
    const int c   = f0 >> 12;
    const int h0  = (f0 >> 6) & 63;
    float a = 0.0f;
#pragma unroll
    for (int r = 0; r < 2; ++r) {
      const float* qrow = qb + (c * HH + (h0 + r)) * WW;
      const float* krow = kb + (c * HH + (h0 + r)) * KWW + p;
      __builtin_prefetch(krow + KWW, 0, 1);     // global_prefetch_b8: next key row
#pragma unroll 16
      for (int w = 0; w < WW; ++w) {
        // lanes 0-15 hit 16 consecutive key columns (coalesced); query broadcasts
        a += fabsf(krow[w] - qrow[w]);
      }
    }
    acc[j] = a;
  }

  // D = A(16x4 partial sums) x B(4x16 ones) + 0  ->  D[m][*] = sum_k acc[m][k]
  v2f A;  A.x = acc[0]; A.y = acc[1];
  v2f Bm; Bm.x = 1.0f;  Bm.y = 1.0f;            // ones matrix: layout-agnostic
  v8f Cm = {};
  v8f D = __builtin_amdgcn_wmma_f32_16x16x4_f32(
      /*neg_a=*/false, A, /*neg_b=*/false, Bm,
      /*c_mod=*/(short)0, Cm, /*reuse_a=*/false, /*reuse_b=*/false);

  // N=0 column: lane 0 holds rows M=0..7 in D[0..7], lane 16 holds M=8..15.
  __shared__ float red[8][16];
  if (lane == 0 || lane == 16) {
    const int mb = khalf * 8;
#pragma unroll
    for (int jj = 0; jj < 8; ++jj) red[wave][mb + jj] = D[jj];
  }
  __syncthreads();

  if (tid < 16) {
    float s = 0.0f;
#pragma unroll
    for (int w2 = 0; w2 < 8; ++w2) s += red[w2][tid];
    partial[((b * NPT + pt) * NFC + fc) * 16 + tid] = s;
  }
}

// Kernel 2: deterministic chunk reduction + mean + min/argmin per batch.
__global__ __launch_bounds__(256)
void PatchSelect_finalize_kernel(const float* __restrict__ partial,
                                 float* __restrict__ out) {
  const int b = blockIdx.x;          // 4 batches
  const int t = threadIdx.x;
  __shared__ float yv[NPT * 16];

  if (t < NPT * 16) {
    const int pt = t >> 4, m = t & 15;
    float s = 0.0f;
#pragma unroll 8
    for (int fc = 0; fc < NFC; ++fc)
      s += partial[((b * NPT + pt) * NFC + fc) * 16 + m];
    yv[t] = (t < PP) ? s * (1.0f / (float)FEAT) : 3.402823466e38f;
  }
  __syncthreads();

  if (t == 0) {
    float best = yv[0];
    int bi = 0;
    for (int pq = 1; pq < PP; ++pq)
      if (yv[pq] < best) { best = yv[pq]; bi = pq; }  // strict '<' = first min
    // Output layout (9 floats): hard_indices[4], P, relevance_maps[4]
    out[b]     = (float)bi;
    out[4]     = (float)PP;   // same value from every block: benign
    out[5 + b] = best;
  }
}

extern "C" void kernel_launch(void* const* d_in, const int* in_sizes, int n_in,
                              void* d_out, int out_size, void* d_ws, size_t ws_size,
                              hipStream_t stream) {
  (void)in_sizes; (void)n_in; (void)out_size; (void)ws_size;
  const float* query = (const float*)d_in[0];
  const float* key   = (const float*)d_in[1];
  float* out     = (float*)d_out;
  float* partial = (float*)d_ws;     // needs 4*9*32*16 floats = 72 KiB

  dim3 grid1(BATCH * NPT * NFC);     // 1152 workgroups
  dim3 blk(256);                     // 8 wave32 waves
  hipLaunchKernelGGL(PatchSelect_partial_kernel, grid1, blk, 0, stream,
                     query, key, partial);
  hipLaunchKernelGGL(PatchSelect_finalize_kernel, dim3(BATCH), blk, 0, stream,
                     partial, out);
}